// Model_2619930051446
// MI455X (gfx1250) — compile-verified
//
#include <hip/hip_runtime.h>
#include <hip/hip_bf16.h>
#include <stdint.h>

#define EXPERTS 128
#define TOPK 8

// Compare-and-swap so that (av,ai) holds the larger value afterwards.
// Values move via v_max_num_f32/v_min_num_f32 (no predicate dependence);
// indices move via one v_cmp_gt_f32 + two v_cndmask.
// Strict '>' keeps the original order on ties (earlier element stays on top).
__device__ __forceinline__ void cas(float& av, int& ai, float& bv, int& bi) {
  bool sw = bv > av;
  float hv = fmaxf(av, bv);
  float lv = fminf(av, bv);
  int   hi = sw ? bi : ai;
  int   li = sw ? ai : bi;
  av = hv; bv = lv; ai = hi; bi = li;
}

// Batcher odd-even mergesort, 8 inputs, 19 comparators, descending.
__device__ __forceinline__ void sort8_desc(float v[8], int ix[8]) {
  cas(v[0],ix[0],v[1],ix[1]); cas(v[2],ix[2],v[3],ix[3]);
  cas(v[4],ix[4],v[5],ix[5]); cas(v[6],ix[6],v[7],ix[7]);
  cas(v[0],ix[0],v[2],ix[2]); cas(v[1],ix[1],v[3],ix[3]);
  cas(v[4],ix[4],v[6],ix[6]); cas(v[5],ix[5],v[7],ix[7]);
  cas(v[1],ix[1],v[2],ix[2]); cas(v[5],ix[5],v[6],ix[6]);
  cas(v[0],ix[0],v[4],ix[4]); cas(v[1],ix[1],v[5],ix[5]);
  cas(v[2],ix[2],v[6],ix[6]); cas(v[3],ix[3],v[7],ix[7]);
  cas(v[2],ix[2],v[4],ix[4]); cas(v[3],ix[3],v[5],ix[5]);
  cas(v[1],ix[1],v[2],ix[2]); cas(v[3],ix[3],v[4],ix[4]);
  cas(v[5],ix[5],v[6],ix[6]);
}

// Bitonic merger, 8 inputs, 12 comparators, descending.
// Input must be (circularly) bitonic -- which max(desc, asc) envelopes are.
__device__ __forceinline__ void bitonic8_desc(float v[8], int ix[8]) {
  cas(v[0],ix[0],v[4],ix[4]); cas(v[1],ix[1],v[5],ix[5]);
  cas(v[2],ix[2],v[6],ix[6]); cas(v[3],ix[3],v[7],ix[7]);
  cas(v[0],ix[0],v[2],ix[2]); cas(v[1],ix[1],v[3],ix[3]);
  cas(v[4],ix[4],v[6],ix[6]); cas(v[5],ix[5],v[7],ix[7]);
  cas(v[0],ix[0],v[1],ix[1]); cas(v[2],ix[2],v[3],ix[3]);
  cas(v[4],ix[4],v[5],ix[5]); cas(v[6],ix[6],v[7],ix[7]);
}

__global__ __launch_bounds__(256) void topk8_softmax_kernel(
    const float* __restrict__ gate,   // [tokens, 128]
    float* __restrict__ out_w,        // [tokens, 8] softmax weights
    float* __restrict__ out_i,        // [tokens, 8] indices (as float)
    int tokens) {
  int token = blockIdx.x * blockDim.x + threadIdx.x;
  if (token >= tokens) return;

  const float4* __restrict__ row =
      (const float4*)(gate + (size_t)token * EXPERTS);

  // gfx1250 prefetch (global_prefetch_b8): pull the back half of the row.
  __builtin_prefetch(((const char*)row) + 256, 0, 1);

  // Running top-8 (sorted descending), seeded from chunk 0.
  float rv[8]; int ri[8];
  {
    float4 a = row[0], b = row[1];
    rv[0]=a.x; rv[1]=a.y; rv[2]=a.z; rv[3]=a.w;
    rv[4]=b.x; rv[5]=b.y; rv[6]=b.z; rv[7]=b.w;
    ri[0]=0; ri[1]=1; ri[2]=2; ri[3]=3; ri[4]=4; ri[5]=5; ri[6]=6; ri[7]=7;
    sort8_desc(rv, ri);
  }

  // 15 more chunks of 8: sort chunk, bitonic top-8 merge with running list.
#pragma unroll 5
  for (int ch = 1; ch < EXPERTS / 8; ++ch) {
    float4 a = row[2 * ch], b = row[2 * ch + 1];
    float cv[8] = {a.x, a.y, a.z, a.w, b.x, b.y, b.z, b.w};
    int ci[8];
#pragma unroll
    for (int c = 0; c < 8; ++c) ci[c] = ch * 8 + c;
    sort8_desc(cv, ci);

    // Top-8 of the union: elementwise max of descending rv and reversed
    // (ascending) cv. Result is circularly bitonic; then 12-CAS resort.
#pragma unroll
    for (int i = 0; i < 8; ++i) {
      bool c = cv[7 - i] > rv[i];        // strict: ties keep earlier index
      float mv = fmaxf(rv[i], cv[7 - i]);
      int   mi = c ? ci[7 - i] : ri[i];
      rv[i] = mv; ri[i] = mi;
    }
    bitonic8_desc(rv, ri);
  }

  // Softmax over the 8 selected logits (rv[0] is the max; exact f32 values).
  float m = rv[0];
  float e[TOPK];
  float s = 0.0f;
#pragma unroll
  for (int j = 0; j < TOPK; ++j) {
    e[j] = __expf(rv[j] - m);   // v_exp_f32 (TRANS32)
    s += e[j];
  }
  float inv = 1.0f / s;

  // Vectorized coalesced stores: 2 x b128 weights, 2 x b128 indices.
  float4* __restrict__ wo = (float4*)(out_w + (size_t)token * TOPK);
  wo[0] = make_float4(e[0] * inv, e[1] * inv, e[2] * inv, e[3] * inv);
  wo[1] = make_float4(e[4] * inv, e[5] * inv, e[6] * inv, e[7] * inv);

  float4* __restrict__ io = (float4*)(out_i + (size_t)token * TOPK);
  io[0] = make_float4((float)ri[0], (float)ri[1], (float)ri[2], (float)ri[3]);
  io[1] = make_float4((float)ri[4], (float)ri[5], (float)ri[6], (float)ri[7]);
}

extern "C" void kernel_launch(void* const* d_in, const int* in_sizes, int n_in,
                              void* d_out, int out_size, void* d_ws, size_t ws_size,
                              hipStream_t stream) {
  (void)n_in; (void)out_size; (void)d_ws; (void)ws_size;
  const float* gate = (const float*)d_in[0];
  int tokens = in_sizes[0] / EXPERTS;        // 262144

  float* out_w = (float*)d_out;                       // [tokens, 8]
  float* out_i = out_w + (size_t)tokens * TOPK;       // [tokens, 8] (as float)

  dim3 block(256);
  dim3 grid((tokens + 255) / 256);
  topk8_softmax_kernel<<<grid, block, 0, stream>>>(gate, out_w, out_i, tokens);
}